// GGNN_1443109011703
// MI455X (gfx1250) — compile-verified
//
#include <hip/hip_runtime.h>
#include <math.h>
#include <stdint.h>

#define B 2
#define N 1024
#define E 4
#define D 64
#define AD 32
#define STEPS 5

typedef __attribute__((ext_vector_type(2))) float v2f;
typedef __attribute__((ext_vector_type(8))) float v8f;

// Full-precision CDNA5 matrix op: D = A(16x4,f32) * B(4x16,f32) + C(16x16,f32)
__device__ __forceinline__ v8f wmma_f32(v2f a, v2f b, v8f c) {
  return __builtin_amdgcn_wmma_f32_16x16x4_f32(
      /*neg_a=*/false, a, /*neg_b=*/false, b,
      /*c_mod=*/(short)0, c, /*reuse_a=*/false, /*reuse_b=*/false);
}

// ---------------------------------------------------------------------------
// proj: in_s[b,e] = h[b] @ W_in[e] + b_in[e]  (and same for out_s / W_oute)
// One wave -> one 16x16 tile of both in_s and out_s (shared A fragments).
// wave-tiles: B*E*(N/16)*(D/16) = 2048
// ---------------------------------------------------------------------------
__global__ __launch_bounds__(128) void proj_kernel(
    const float* __restrict__ h,
    const float* __restrict__ W_in,  const float* __restrict__ b_in,
    const float* __restrict__ W_out, const float* __restrict__ b_out,
    float* __restrict__ in_s, float* __restrict__ out_s) {
  int wid  = blockIdx.x * 4 + (threadIdx.x >> 5);
  int lane = threadIdx.x & 31;
  int t = wid;
  int nt = t & 3;  t >>= 2;
  int mt = t & 63; t >>= 6;
  int e  = t & 3;  int b = t >> 2;
  int lm = lane & 15, lh = lane >> 4;

  int row = mt * 16 + lm;   // node index (A row)
  int col = nt * 16 + lm;   // feature index (B col / store col)
  const float* __restrict__ Arow = h + ((size_t)(b * N + row)) * D;
  const float* __restrict__ Bi   = W_in  + (size_t)e * D * D;
  const float* __restrict__ Bo   = W_out + (size_t)e * D * D;

  v8f ci = {}; v8f co = {};
  #pragma unroll
  for (int k0 = 0; k0 < D; k0 += 4) {
    int kk = k0 + 2 * lh;
    v2f a;  a.x  = Arow[kk];           a.y  = Arow[kk + 1];
    v2f bi; bi.x = Bi[kk * D + col];   bi.y = Bi[(kk + 1) * D + col];
    v2f bo; bo.x = Bo[kk * D + col];   bo.y = Bo[(kk + 1) * D + col];
    ci = wmma_f32(a, bi, ci);
    co = wmma_f32(a, bo, co);
  }
  float biasi = b_in[e * D + col];
  float biaso = b_out[e * D + col];
  float* __restrict__ pin  = in_s  + ((size_t)(b * E + e) * N) * D;
  float* __restrict__ pout = out_s + ((size_t)(b * E + e) * N) * D;
  #pragma unroll
  for (int v = 0; v < 8; ++v) {
    int m = mt * 16 + v + 8 * lh;
    pin [(size_t)m * D + col] = ci[v] + biasi;
    pout[(size_t)m * D + col] = co[v] + biaso;
  }
}

// ---------------------------------------------------------------------------
// agg: a_in[b]  = A_in[b]  @ in_s[b]   (sel=0)
//      a_out[b] = A_out[b] @ out_s[b]  (sel=1)
// (1024 x 4096) @ (4096 x 64).
// Block = 8 waves, one (b,sel) per block; all waves share the B operand,
// staged chunk-wise into LDS with gfx1250 async global->LDS copies.
// One wave computes a full 16x64 row-block (4 accumulators), so the 64MB
// adjacency is streamed exactly once per step.
// ---------------------------------------------------------------------------
#define KC 128          // K-chunk staged in LDS
#define BPITCH 80       // 64 + 16 pad: halves of the wave hit disjoint banks

__global__ __launch_bounds__(256) void agg_kernel(
    const float* __restrict__ adj,
    const float* __restrict__ in_s, const float* __restrict__ out_s,
    float* __restrict__ a_in, float* __restrict__ a_out) {
  __shared__ __align__(16) float Bs[KC * BPITCH];   // 40 KB (of 320 KB/WGP)

  int tid  = threadIdx.x;
  int wave = tid >> 5;
  int lane = tid & 31;
  int lm = lane & 15, lh = lane >> 4;

  int gid   = blockIdx.x;        // 0..31
  int combo = gid >> 3;          // 0..3 : {sel,b}
  int sel = combo >> 1, b = combo & 1;
  int mt  = (gid & 7) * 8 + wave;   // row-tile 0..63

  const int ldA = 2 * N * E;     // 8192
  int row = mt * 16 + lm;
  const float* __restrict__ Arow =
      adj + (size_t)(b * N + row) * ldA + (size_t)sel * (N * E);
  const float* __restrict__ Bsrc =
      (sel ? out_s : in_s) + (size_t)b * E * N * D;

  // cooperative async staging: 256 threads x 16B; 16 rows/pass, 8 passes/chunk
  int crow = tid >> 4;           // 0..15
  int ccol = (tid & 15) * 4;     // float column 0,4,..,60

  v8f c0 = {}, c1 = {}, c2 = {}, c3 = {};

  for (int k0 = 0; k0 < N * E; k0 += KC) {
    #pragma unroll
    for (int p = 0; p < KC / 16; ++p) {
      int kr = p * 16 + crow;
      // Low 32 bits of the flat address of an LDS location ARE the raw LDS
      // byte offset (ISA 10.2 aperture rules).  Deriving the operand from
      // &Bs[...] also makes Bs escape into the asm, so the compiler must
      // treat the async op as writing it (keeps the ds_load fragment reads).
      unsigned lds_off =
          (unsigned)(uintptr_t)(&Bs[kr * BPITCH + ccol]);
      const float* g = Bsrc + (size_t)(k0 + kr) * D + ccol;
      asm volatile("global_load_async_to_lds_b128 %0, %1, off"
                   :: "v"(lds_off), "v"((unsigned long long)(uintptr_t)g)
                   : "memory");
    }
    asm volatile("s_wait_asynccnt 0x0" ::: "memory");
    __syncthreads();

    #pragma unroll 4
    for (int kq = 0; kq < KC; kq += 4) {
      int kk = kq + 2 * lh;
      v2f a;
      a.x = Arow[k0 + kk];
      a.y = Arow[k0 + kk + 1];
      const float* __restrict__ Br = &Bs[kk * BPITCH + lm];
      v2f b0; b0.x = Br[ 0]; b0.y = Br[BPITCH +  0];
      v2f b1; b1.x = Br[16]; b1.y = Br[BPITCH + 16];
      v2f b2; b2.x = Br[32]; b2.y = Br[BPITCH + 32];
      v2f b3; b3.x = Br[48]; b3.y = Br[BPITCH + 48];
      c0 = wmma_f32(a, b0, c0);
      c1 = wmma_f32(a, b1, c1);
      c2 = wmma_f32(a, b2, c2);
      c3 = wmma_f32(a, b3, c3);
    }
    __syncthreads();
  }

  float* __restrict__ dst = (sel ? a_out : a_in) + (size_t)b * N * D;
  #pragma unroll
  for (int v = 0; v < 8; ++v) {
    int m = mt * 16 + v + 8 * lh;
    float* __restrict__ drow = dst + (size_t)m * D + lm;
    drow[ 0] = c0[v];
    drow[16] = c1[v];
    drow[32] = c2[v];
    drow[48] = c3[v];
  }
}

// ---------------------------------------------------------------------------
// gates: a = [a_in | a_out | h] (K=192)
//        z = sigmoid(a @ W_z + b_z); r = sigmoid(a @ W_r + b_r); rh = r*h
// wave-tiles: B*(N/16)*(D/16) = 512
// ---------------------------------------------------------------------------
__global__ __launch_bounds__(128) void gates_kernel(
    const float* __restrict__ a_in, const float* __restrict__ a_out,
    const float* __restrict__ h,
    const float* __restrict__ W_z, const float* __restrict__ b_z,
    const float* __restrict__ W_r, const float* __restrict__ b_r,
    float* __restrict__ z_out, float* __restrict__ rh_out) {
  int wid  = blockIdx.x * 4 + (threadIdx.x >> 5);
  int lane = threadIdx.x & 31;
  int t = wid;
  int nt = t & 3;  t >>= 2;
  int mt = t & 63; t >>= 6;
  int b  = t;
  int lm = lane & 15, lh = lane >> 4;

  int row = mt * 16 + lm;
  int col = nt * 16 + lm;
  size_t base = (size_t)(b * N) * D;
  const float* segs[3] = { a_in + base, a_out + base, h + base };

  v8f cz = {}; v8f cr = {};
  #pragma unroll
  for (int k0 = 0; k0 < 3 * D; k0 += 4) {
    const float* __restrict__ S = segs[k0 >> 6];   // 64-aligned segments
    int ks = (k0 & 63) + 2 * lh;                   // offset within segment
    int kw = k0 + 2 * lh;                          // row into W (0..191)
    v2f a;  a.x  = S[(size_t)row * D + ks];  a.y  = S[(size_t)row * D + ks + 1];
    v2f bz; bz.x = W_z[kw * D + col];        bz.y = W_z[(kw + 1) * D + col];
    v2f br; br.x = W_r[kw * D + col];        br.y = W_r[(kw + 1) * D + col];
    cz = wmma_f32(a, bz, cz);
    cr = wmma_f32(a, br, cr);
  }
  float bzv = b_z[col], brv = b_r[col];
  #pragma unroll
  for (int v = 0; v < 8; ++v) {
    int m = mt * 16 + v + 8 * lh;
    size_t idx = (size_t)(b * N + m) * D + col;
    float zz = 1.0f / (1.0f + expf(-(cz[v] + bzv)));
    float rr = 1.0f / (1.0f + expf(-(cr[v] + brv)));
    z_out[idx]  = zz;
    rh_out[idx] = rr * h[idx];
  }
}

// ---------------------------------------------------------------------------
// hnext: h_hat = tanh([a_in | a_out | rh] @ W_h + b_h)
//        h     = (1-z)*h + z*h_hat      (in place on h)
// wave-tiles: 512
// ---------------------------------------------------------------------------
__global__ __launch_bounds__(128) void hnext_kernel(
    const float* __restrict__ a_in, const float* __restrict__ a_out,
    const float* __restrict__ rh,
    const float* __restrict__ W_h, const float* __restrict__ b_h,
    const float* __restrict__ z_in, float* __restrict__ h) {
  int wid  = blockIdx.x * 4 + (threadIdx.x >> 5);
  int lane = threadIdx.x & 31;
  int t = wid;
  int nt = t & 3;  t >>= 2;
  int mt = t & 63; t >>= 6;
  int b  = t;
  int lm = lane & 15, lh = lane >> 4;

  int row = mt * 16 + lm;
  int col = nt * 16 + lm;
  size_t base = (size_t)(b * N) * D;
  const float* segs[3] = { a_in + base, a_out + base, rh + base };

  v8f c = {};
  #pragma unroll
  for (int k0 = 0; k0 < 3 * D; k0 += 4) {
    const float* __restrict__ S = segs[k0 >> 6];
    int ks = (k0 & 63) + 2 * lh;
    int kw = k0 + 2 * lh;
    v2f a;  a.x  = S[(size_t)row * D + ks];  a.y  = S[(size_t)row * D + ks + 1];
    v2f bh; bh.x = W_h[kw * D + col];        bh.y = W_h[(kw + 1) * D + col];
    c = wmma_f32(a, bh, c);
  }
  float bhv = b_h[col];
  #pragma unroll
  for (int v = 0; v < 8; ++v) {
    int m = mt * 16 + v + 8 * lh;
    size_t idx = (size_t)(b * N + m) * D + col;
    float hh = tanhf(c[v] + bhv);
    float zz = z_in[idx];
    h[idx] = (1.0f - zz) * h[idx] + zz * hh;
  }
}

// ---------------------------------------------------------------------------
// output head: o[b,n] = tanh([h|annot] @ W_o1 + b_o1) @ W_o2 + b_o2
// Tiny (2048 nodes x 6k MACs) -> plain VALU.
// ---------------------------------------------------------------------------
__global__ __launch_bounds__(256) void out_kernel(
    const float* __restrict__ h, const float* __restrict__ annot,
    const float* __restrict__ W_o1, const float* __restrict__ b_o1,
    const float* __restrict__ W_o2, const float* __restrict__ b_o2,
    float* __restrict__ out) {
  int i = blockIdx.x * blockDim.x + threadIdx.x;
  if (i >= B * N) return;
  float x[D + AD];
  #pragma unroll
  for (int k = 0; k < D; ++k)  x[k]     = h[(size_t)i * D + k];
  #pragma unroll
  for (int k = 0; k < AD; ++k) x[D + k] = annot[(size_t)i * AD + k];
  float acc = 0.0f;
  for (int f = 0; f < D; ++f) {
    float s = b_o1[f];
    #pragma unroll 8
    for (int k = 0; k < D + AD; ++k) s += x[k] * W_o1[k * D + f];
    acc += tanhf(s) * W_o2[f];
  }
  out[i] = acc + b_o2[0];
}

// ---------------------------------------------------------------------------
extern "C" void kernel_launch(void* const* d_in, const int* in_sizes, int n_in,
                              void* d_out, int out_size, void* d_ws, size_t ws_size,
                              hipStream_t stream) {
  const float* init_h = (const float*)d_in[0];
  const float* annot  = (const float*)d_in[1];
  const float* adj    = (const float*)d_in[2];
  const float* W_in   = (const float*)d_in[3];
  const float* b_in   = (const float*)d_in[4];
  const float* W_oute = (const float*)d_in[5];
  const float* b_oute = (const float*)d_in[6];
  const float* W_z    = (const float*)d_in[7];
  const float* b_z    = (const float*)d_in[8];
  const float* W_r    = (const float*)d_in[9];
  const float* b_r    = (const float*)d_in[10];
  const float* W_h    = (const float*)d_in[11];
  const float* b_h    = (const float*)d_in[12];
  const float* W_o1   = (const float*)d_in[13];
  const float* b_o1   = (const float*)d_in[14];
  const float* W_o2   = (const float*)d_in[15];
  const float* b_o2   = (const float*)d_in[16];

  float* ws    = (float*)d_ws;
  float* in_s  = ws;                         // B*E*N*D = 524288
  float* out_s = in_s  + (size_t)B * E * N * D;
  float* a_in  = out_s + (size_t)B * E * N * D;  // B*N*D = 131072
  float* a_out = a_in  + (size_t)B * N * D;
  float* hbuf  = a_out + (size_t)B * N * D;
  float* zbuf  = hbuf  + (size_t)B * N * D;
  float* rhbuf = zbuf  + (size_t)B * N * D;

  hipMemcpyAsync(hbuf, init_h, (size_t)B * N * D * sizeof(float),
                 hipMemcpyDeviceToDevice, stream);

  for (int s = 0; s < STEPS; ++s) {
    proj_kernel <<<512, 128, 0, stream>>>(hbuf, W_in, b_in, W_oute, b_oute,
                                          in_s, out_s);
    agg_kernel  <<<32, 256, 0, stream>>>(adj, in_s, out_s, a_in, a_out);
    gates_kernel<<<128, 128, 0, stream>>>(a_in, a_out, hbuf, W_z, b_z, W_r, b_r,
                                          zbuf, rhbuf);
    hnext_kernel<<<128, 128, 0, stream>>>(a_in, a_out, rhbuf, W_h, b_h, zbuf,
                                          hbuf);
  }
  out_kernel<<<(B * N + 255) / 256, 256, 0, stream>>>(
      hbuf, annot, W_o1, b_o1, W_o2, b_o2, (float*)d_out);
}